// EfficientSpatialContextNet_49031346651469
// MI455X (gfx1250) — compile-verified
//
#include <hip/hip_runtime.h>
#include <hip/hip_bf16.h>

typedef __attribute__((ext_vector_type(2))) float v2f;
typedef __attribute__((ext_vector_type(8))) float v8f;

#define CH    768
#define HW    4096
#define NOFF  28
#define KTOT  796   // 768 + 28

// ---------------------------------------------------------------------------
// Kernel 1: per-pixel inverse L2 norm over channels.
// ---------------------------------------------------------------------------
__global__ void escn_norm_kernel(const float* __restrict__ feat,
                                 float* __restrict__ invn) {
    int idx = blockIdx.x * blockDim.x + threadIdx.x;   // 0..32767
    int b = idx >> 12;
    int p = idx & (HW - 1);
    const float* base = feat + ((size_t)b * CH) * HW + p;
    float s = 0.0f;
#pragma unroll 4
    for (int c = 0; c < CH; ++c) {
        float v = base[(size_t)c * HW];
        s = fmaf(v, v, s);
    }
    float n = sqrtf(s);
    invn[idx] = 1.0f / fmaxf(n, 1e-12f);
}

// ---------------------------------------------------------------------------
// Kernel 2: 28-offset cost volume. One thread per (b, pixel); accumulates all
// 28 shifted dot products in one channel sweep. Out-of-bounds offsets are
// clamped to the center pixel and zeroed at the end (branchless inner loop).
// ---------------------------------------------------------------------------
__global__ void escn_corr_kernel(const float* __restrict__ feat,
                                 const float* __restrict__ invn,
                                 float* __restrict__ corr) {
    int idx = blockIdx.x * blockDim.x + threadIdx.x;   // 0..32767
    int b = idx >> 12;
    int p = idx & (HW - 1);
    int h = p >> 6;
    int w = p & 63;

    int   sp[NOFF];
    float vf[NOFF];
#pragma unroll
    for (int o = 0; o < NOFF; ++o) {
        int r, cc;
        if      (o <  7) { r = o;      cc = o;            }  // main diagonal
        else if (o < 14) { r = o - 7;  cc = 3;            }  // vertical
        else if (o < 21) { r = o - 14; cc = 6 - (o - 14); }  // anti-diagonal
        else             { r = 3;      cc = o - 21;       }  // horizontal
        int sh = h + r - 3;
        int sw = w + cc - 3;
        bool ok = (sh >= 0) && (sh < 64) && (sw >= 0) && (sw < 64);
        sp[o] = ok ? (sh * 64 + sw) : p;
        vf[o] = ok ? 1.0f : 0.0f;
    }

    float acc[NOFF];
#pragma unroll
    for (int o = 0; o < NOFF; ++o) acc[o] = 0.0f;

    const float* base = feat + ((size_t)b * CH) * HW;
    for (int c = 0; c < CH; ++c) {
        const float* row = base + (size_t)c * HW;
        float center = row[p];
#pragma unroll
        for (int o = 0; o < NOFF; ++o)
            acc[o] = fmaf(row[sp[o]], center, acc[o]);
    }

    const float* ib = invn + b * HW;
    float ic = ib[p];
#pragma unroll
    for (int o = 0; o < NOFF; ++o)
        corr[((size_t)b * NOFF + o) * HW + p] = acc[o] * vf[o] * ic * ib[sp[o]];
}

// ---------------------------------------------------------------------------
// Kernel 3: 1x1 conv as fp32 WMMA GEMM.
//   Out[b][o][m] = sum_k W[o][k] * X[b][k][m] + bias[o]
//   X rows 0..767 = feature[b], rows 768..795 = corr[b]. Both [k][m], stride HW.
// Each wave computes a 32(o) x 64(m) strip: 8 accumulator frags (64 VGPRs).
// Per k-step: 4 A loads + 8 B loads -> 8 WMMAs (1.5 loads/WMMA).
// K loop split at the frag-aligned k=768 boundary so there is no per-iteration
// pointer select (avoids the accumulator register shuffles seen previously).
// Fragment layouts per CDNA5 ISA 7.12.2 (wave32):
//   A 16x4 : lane%16 = M row,  lane/16 selects K pair {0,1} vs {2,3}; vgpr = K%2
//   B 4x16 : lane%16 = N col,  lane/16 selects K pair;              vgpr = K%2
//   D 16x16: lane%16 = N col,  M = vgpr + 8*(lane/16)
// ---------------------------------------------------------------------------
__global__ void escn_gemm_kernel(const float* __restrict__ feat,
                                 const float* __restrict__ corr,
                                 const float* __restrict__ Wm,
                                 const float* __restrict__ bias,
                                 float* __restrict__ out) {
    const int lane  = threadIdx.x & 31;
    const int wid   = threadIdx.x >> 5;          // 0..7 waves per block
    const int mtile = blockIdx.x * 8 + wid;      // 0..63  (64 m-strips of 64)
    const int m0    = mtile * 64;
    const int o0    = blockIdx.y * 32;           // 24 o-tiles of 32
    const int b     = blockIdx.z;                // 8 batches

    const int col   = lane & 15;
    const int khalf = lane >> 4;                 // 0: K{0,1}, 1: K{2,3}

    v8f acc00 = {}, acc01 = {}, acc02 = {}, acc03 = {};   // o rows 0..15
    v8f acc10 = {}, acc11 = {}, acc12 = {}, acc13 = {};   // o rows 16..31

    const float* fbase = feat + ((size_t)b * CH)   * HW + m0 + col;
    const float* cbase = corr + ((size_t)b * NOFF) * HW + m0 + col;
    const float* wrow0 = Wm + (size_t)(o0 + col) * KTOT + 2 * khalf;
    const float* wrow1 = wrow0 + (size_t)16 * KTOT;

    // ---- K part 1: feature rows (192 frags) ----
    for (int k = 0; k < CH; k += 4) {
        v2f a0 = *(const v2f*)(wrow0 + k);
        v2f a1 = *(const v2f*)(wrow1 + k);

        const float* xk = fbase + ((size_t)k + 2 * khalf) * HW;
        v2f b0, b1, b2, b3;
        b0.x = xk[0];  b0.y = xk[HW + 0];
        b1.x = xk[16]; b1.y = xk[HW + 16];
        b2.x = xk[32]; b2.y = xk[HW + 32];
        b3.x = xk[48]; b3.y = xk[HW + 48];

        acc00 = __builtin_amdgcn_wmma_f32_16x16x4_f32(false, a0, false, b0, (short)0, acc00, false, false);
        acc01 = __builtin_amdgcn_wmma_f32_16x16x4_f32(false, a0, false, b1, (short)0, acc01, false, false);
        acc02 = __builtin_amdgcn_wmma_f32_16x16x4_f32(false, a0, false, b2, (short)0, acc02, false, false);
        acc03 = __builtin_amdgcn_wmma_f32_16x16x4_f32(false, a0, false, b3, (short)0, acc03, false, false);
        acc10 = __builtin_amdgcn_wmma_f32_16x16x4_f32(false, a1, false, b0, (short)0, acc10, false, false);
        acc11 = __builtin_amdgcn_wmma_f32_16x16x4_f32(false, a1, false, b1, (short)0, acc11, false, false);
        acc12 = __builtin_amdgcn_wmma_f32_16x16x4_f32(false, a1, false, b2, (short)0, acc12, false, false);
        acc13 = __builtin_amdgcn_wmma_f32_16x16x4_f32(false, a1, false, b3, (short)0, acc13, false, false);
    }

    // ---- K part 2: corr rows (7 frags) ----
    for (int k = 0; k < NOFF; k += 4) {
        v2f a0 = *(const v2f*)(wrow0 + CH + k);
        v2f a1 = *(const v2f*)(wrow1 + CH + k);

        const float* xk = cbase + ((size_t)k + 2 * khalf) * HW;
        v2f b0, b1, b2, b3;
        b0.x = xk[0];  b0.y = xk[HW + 0];
        b1.x = xk[16]; b1.y = xk[HW + 16];
        b2.x = xk[32]; b2.y = xk[HW + 32];
        b3.x = xk[48]; b3.y = xk[HW + 48];

        acc00 = __builtin_amdgcn_wmma_f32_16x16x4_f32(false, a0, false, b0, (short)0, acc00, false, false);
        acc01 = __builtin_amdgcn_wmma_f32_16x16x4_f32(false, a0, false, b1, (short)0, acc01, false, false);
        acc02 = __builtin_amdgcn_wmma_f32_16x16x4_f32(false, a0, false, b2, (short)0, acc02, false, false);
        acc03 = __builtin_amdgcn_wmma_f32_16x16x4_f32(false, a0, false, b3, (short)0, acc03, false, false);
        acc10 = __builtin_amdgcn_wmma_f32_16x16x4_f32(false, a1, false, b0, (short)0, acc10, false, false);
        acc11 = __builtin_amdgcn_wmma_f32_16x16x4_f32(false, a1, false, b1, (short)0, acc11, false, false);
        acc12 = __builtin_amdgcn_wmma_f32_16x16x4_f32(false, a1, false, b2, (short)0, acc12, false, false);
        acc13 = __builtin_amdgcn_wmma_f32_16x16x4_f32(false, a1, false, b3, (short)0, acc13, false, false);
    }

    // Store + bias: lane L, vgpr v -> o = o0 + 16*ot + v + 8*khalf,
    //                                 m = m0 + 16*t + col
    float* outb = out + ((size_t)b * CH) * HW + m0 + col;
#pragma unroll
    for (int v = 0; v < 8; ++v) {
        int oA = o0 + v + 8 * khalf;
        int oB = oA + 16;
        float bsA = bias[oA];
        float bsB = bias[oB];
        float* rowA = outb + (size_t)oA * HW;
        float* rowB = outb + (size_t)oB * HW;
        rowA[0]  = acc00[v] + bsA;
        rowA[16] = acc01[v] + bsA;
        rowA[32] = acc02[v] + bsA;
        rowA[48] = acc03[v] + bsA;
        rowB[0]  = acc10[v] + bsB;
        rowB[16] = acc11[v] + bsB;
        rowB[32] = acc12[v] + bsB;
        rowB[48] = acc13[v] + bsB;
    }
}

// ---------------------------------------------------------------------------
extern "C" void kernel_launch(void* const* d_in, const int* in_sizes, int n_in,
                              void* d_out, int out_size, void* d_ws, size_t ws_size,
                              hipStream_t stream) {
    const float* feature = (const float*)d_in[0];   // [8,768,64,64]
    const float* conv_w  = (const float*)d_in[1];   // [768,796]
    const float* conv_b  = (const float*)d_in[2];   // [768]
    float* out = (float*)d_out;                     // [8,768,64,64]

    float* invn = (float*)d_ws;                     // 32768 floats
    float* corr = invn + 32768;                     // 8*28*4096 floats (3.5 MB)

    escn_norm_kernel<<<128, 256, 0, stream>>>(feature, invn);
    escn_corr_kernel<<<128, 256, 0, stream>>>(feature, invn, corr);
    escn_gemm_kernel<<<dim3(8, 24, 8), 256, 0, stream>>>(feature, corr,
                                                         conv_w, conv_b, out);
}